// CausalGNN_50474455662936
// MI455X (gfx1250) — compile-verified
//
#include <hip/hip_runtime.h>
#include <hip/hip_bf16.h>

// GAT (3 layers, PyG GATConv semantics) for MI455X / gfx1250.
//  - Dense per-node transforms use V_WMMA_F32_16X16X4_F32 (wave32 matrix path).
//  - Edge softmax/scatter phases use f32 L2 atomics (all hot arrays fit in 192MB L2).

typedef __attribute__((ext_vector_type(2))) float v2f;
typedef __attribute__((ext_vector_type(8))) float v8f;

#define BLK 256

// ---------------------------------------------------------------- utilities

__global__ void fill_kernel(float* __restrict__ p, float v, long long n) {
    long long i = (long long)blockIdx.x * blockDim.x + threadIdx.x;
    if (i < n) p[i] = v;
}

__device__ __forceinline__ void atomicMaxF(float* addr, float val) {
    // monotonic float-max via integer views (works for mixed signs, init = -1e30)
    if (val >= 0.0f) {
        atomicMax((int*)addr, __float_as_int(val));
    } else {
        atomicMin((unsigned int*)addr, __float_as_uint(val));
    }
}

// ------------------------------------------------- layer 1: x[N,3] @ W1[3,32]
// Fused: also computes per-head attention dots al_s, al_d (H=2, C=16).

__global__ void linear1_kernel(const float* __restrict__ x,
                               const float* __restrict__ W1,
                               const float* __restrict__ a_src,
                               const float* __restrict__ a_dst,
                               float* __restrict__ h,
                               float* __restrict__ als,
                               float* __restrict__ ald, int N) {
    int n = blockIdx.x * blockDim.x + threadIdx.x;
    if (n >= N) return;
    float x0 = x[3 * n + 0], x1 = x[3 * n + 1], x2 = x[3 * n + 2];
    float acc_s0 = 0.f, acc_s1 = 0.f, acc_d0 = 0.f, acc_d1 = 0.f;
#pragma unroll
    for (int c = 0; c < 32; ++c) {
        float v = x0 * W1[c] + x1 * W1[32 + c] + x2 * W1[64 + c];
        h[(long long)n * 32 + c] = v;
        int head = c >> 4, cc = c & 15;
        float as = a_src[head * 16 + cc];
        float ad = a_dst[head * 16 + cc];
        if (head == 0) { acc_s0 += v * as; acc_d0 += v * ad; }
        else           { acc_s1 += v * as; acc_d1 += v * ad; }
    }
    als[2 * n + 0] = acc_s0; als[2 * n + 1] = acc_s1;
    ald[2 * n + 0] = acc_d0; ald[2 * n + 1] = acc_d1;
}

// ------------------------------------- WMMA linear: X[N,32] @ W[32,32] -> [N,32]
// One wave computes a 16-row tile; K=32 via 8 x V_WMMA_F32_16X16X4_F32 per col-tile.

__global__ void wmma_linear32_kernel(const float* __restrict__ X,
                                     const float* __restrict__ W,
                                     float* __restrict__ Hout,
                                     int nTiles, int N) {
    int lane = threadIdx.x & 31;
    int wave = threadIdx.x >> 5;
    int tile = blockIdx.x * (blockDim.x >> 5) + wave;   // wave-uniform
    if (tile >= nTiles) return;                          // wave-uniform exit
    int node0 = tile * 16;
    int row   = node0 + (lane & 15);
    int rowc  = row < N ? row : N - 1;
    int koff  = (lane >> 4) << 1;                        // lanes 16-31 hold K+2,K+3
    int col   = lane & 15;

    v8f c0 = {}; v8f c1 = {};
#pragma unroll
    for (int k0 = 0; k0 < 32; k0 += 4) {
        int kb = k0 + koff;
        v2f a;  a.x  = X[(long long)rowc * 32 + kb];     a.y  = X[(long long)rowc * 32 + kb + 1];
        v2f b0; b0.x = W[kb * 32 + col];                 b0.y = W[(kb + 1) * 32 + col];
        v2f b1; b1.x = W[kb * 32 + 16 + col];            b1.y = W[(kb + 1) * 32 + 16 + col];
        c0 = __builtin_amdgcn_wmma_f32_16x16x4_f32(false, a, false, b0, (short)0, c0, false, false);
        c1 = __builtin_amdgcn_wmma_f32_16x16x4_f32(false, a, false, b1, (short)0, c1, false, false);
    }
    // C/D layout: VGPR i -> M=i (lanes 0-15) or M=8+i (lanes 16-31), N = lane&15
    int mbase = node0 + ((lane >> 4) << 3);
#pragma unroll
    for (int i = 0; i < 8; ++i) {
        int r = mbase + i;
        if (r < N) {
            Hout[(long long)r * 32 + col]      = c0[i];
            Hout[(long long)r * 32 + 16 + col] = c1[i];
        }
    }
}

// ------------------- WMMA linear: X[N,32] @ Wpad[32,16] -> [N,8] (cols 8..15 zero)

__global__ void wmma_linear8_kernel(const float* __restrict__ X,
                                    const float* __restrict__ Wp,   // 32x16 padded
                                    float* __restrict__ Hout,       // stride 8
                                    int nTiles, int N) {
    int lane = threadIdx.x & 31;
    int wave = threadIdx.x >> 5;
    int tile = blockIdx.x * (blockDim.x >> 5) + wave;
    if (tile >= nTiles) return;
    int node0 = tile * 16;
    int row   = node0 + (lane & 15);
    int rowc  = row < N ? row : N - 1;
    int koff  = (lane >> 4) << 1;
    int col   = lane & 15;

    v8f c0 = {};
#pragma unroll
    for (int k0 = 0; k0 < 32; k0 += 4) {
        int kb = k0 + koff;
        v2f a;  a.x  = X[(long long)rowc * 32 + kb]; a.y = X[(long long)rowc * 32 + kb + 1];
        v2f b0; b0.x = Wp[kb * 16 + col];            b0.y = Wp[(kb + 1) * 16 + col];
        c0 = __builtin_amdgcn_wmma_f32_16x16x4_f32(false, a, false, b0, (short)0, c0, false, false);
    }
    int mbase = node0 + ((lane >> 4) << 3);
#pragma unroll
    for (int i = 0; i < 8; ++i) {
        int r = mbase + i;
        if (r < N && col < 8) Hout[(long long)r * 8 + col] = c0[i];
    }
}

__global__ void padw3_kernel(const float* __restrict__ W3, float* __restrict__ Wp) {
    int i = blockIdx.x * blockDim.x + threadIdx.x;
    if (i >= 32 * 16) return;
    int r = i >> 4, c = i & 15;
    Wp[i] = (c < 8) ? W3[r * 8 + c] : 0.0f;
}

// --------------------------------------- attention dots for layers 2/3

__global__ void attn_kernel(const float* __restrict__ h,
                            const float* __restrict__ a_src,
                            const float* __restrict__ a_dst,
                            float* __restrict__ als, float* __restrict__ ald,
                            int N, int H, int C) {
    int t = blockIdx.x * blockDim.x + threadIdx.x;
    if (t >= N * H) return;
    int n = t / H, hd = t % H;
    const float* hp = h + (long long)n * H * C + hd * C;
    float s = 0.f, d = 0.f;
    for (int c = 0; c < C; ++c) {
        float v = hp[c];
        s += v * a_src[hd * C + c];
        d += v * a_dst[hd * C + c];
    }
    als[t] = s; ald[t] = d;
}

// --------------------------------------------------------- edge phases

__global__ void edge_logit_max_kernel(const int* __restrict__ ei, int E, int N,
                                      const float* __restrict__ als,
                                      const float* __restrict__ ald,
                                      float* __restrict__ m,
                                      float* __restrict__ elog, int H) {
    int e = blockIdx.x * blockDim.x + threadIdx.x;
    if (e >= E + N) return;
    int s_, d_;
    if (e < E) { s_ = ei[e]; d_ = ei[E + e]; } else { s_ = d_ = e - E; }
    for (int h = 0; h < H; ++h) {
        float l = als[s_ * H + h] + ald[d_ * H + h];
        l = l > 0.f ? l : 0.2f * l;                 // leaky_relu(0.2)
        elog[(long long)e * H + h] = l;
        atomicMaxF(&m[d_ * H + h], l);
    }
}

__global__ void edge_exp_sum_kernel(const int* __restrict__ ei, int E, int N,
                                    const float* __restrict__ m,
                                    float* __restrict__ ssum,
                                    float* __restrict__ elog, int H) {
    int e = blockIdx.x * blockDim.x + threadIdx.x;
    if (e >= E + N) return;
    int d_ = (e < E) ? ei[E + e] : (e - E);
    for (int h = 0; h < H; ++h) {
        float ex = __expf(elog[(long long)e * H + h] - m[d_ * H + h]);
        elog[(long long)e * H + h] = ex;
        atomicAdd(&ssum[d_ * H + h], ex);
    }
}

// one thread per (edge, 4-channel group): gather h[src], scatter alpha-weighted msg
__global__ void edge_scatter_kernel(const int* __restrict__ ei, int E, int N,
                                    const float* __restrict__ h,
                                    const float* __restrict__ elog,
                                    const float* __restrict__ ssum,
                                    float* __restrict__ agg, int H, int C) {
    int HC = H * C;
    int groups = HC >> 2;
    long long tid = (long long)blockIdx.x * blockDim.x + threadIdx.x;
    long long total = (long long)(E + N) * groups;
    if (tid >= total) return;
    int e = (int)(tid / groups);
    int q = (int)(tid % groups);
    int s_, d_;
    if (e < E) { s_ = ei[e]; d_ = ei[E + e]; } else { s_ = d_ = e - E; }
    int ch = q << 2;
    int head = ch / C;
    float alpha = elog[(long long)e * H + head] / (ssum[d_ * H + head] + 1e-16f);
    const float4 hv = *(const float4*)(h + (long long)s_ * HC + ch);
    float* ap = agg + (long long)d_ * HC + ch;
    atomicAdd(ap + 0, hv.x * alpha);
    atomicAdd(ap + 1, hv.y * alpha);
    atomicAdd(ap + 2, hv.z * alpha);
    atomicAdd(ap + 3, hv.w * alpha);
}

// ----------------------------------------------- bias + ELU

__global__ void finalize_kernel(const float* __restrict__ agg,
                                const float* __restrict__ bias,
                                float* __restrict__ out, int N, int HC) {
    long long i = (long long)blockIdx.x * blockDim.x + threadIdx.x;
    if (i >= (long long)N * HC) return;
    float v = agg[i] + bias[(int)(i % HC)];
    out[i] = v > 0.f ? v : (__expf(v) - 1.f);
}

// ----------------------------------------------- final projection h3[N,8]@Wo[8,1]+bo

__global__ void output_kernel(const float* __restrict__ h,
                              const float* __restrict__ Wo,
                              const float* __restrict__ bo,
                              float* __restrict__ out, int N) {
    int n = blockIdx.x * blockDim.x + threadIdx.x;
    if (n >= N) return;
    float acc = bo[0];
#pragma unroll
    for (int c = 0; c < 8; ++c) acc += h[(long long)n * 8 + c] * Wo[c];
    out[n] = acc;
}

// ================================================================ launcher

static inline int nblk(long long n) { return (int)((n + BLK - 1) / BLK); }

extern "C" void kernel_launch(void* const* d_in, const int* in_sizes, int n_in,
                              void* d_out, int out_size, void* d_ws, size_t ws_size,
                              hipStream_t stream) {
    const float* x   = (const float*)d_in[0];
    const int*   ei  = (const int*)  d_in[1];
    const float* W1  = (const float*)d_in[2];
    const float* as1 = (const float*)d_in[3];
    const float* ad1 = (const float*)d_in[4];
    const float* b1  = (const float*)d_in[5];
    const float* W2  = (const float*)d_in[6];
    const float* as2 = (const float*)d_in[7];
    const float* ad2 = (const float*)d_in[8];
    const float* b2  = (const float*)d_in[9];
    const float* W3  = (const float*)d_in[10];
    const float* as3 = (const float*)d_in[11];
    const float* ad3 = (const float*)d_in[12];
    const float* b3  = (const float*)d_in[13];
    const float* Wo  = (const float*)d_in[14];
    const float* bo  = (const float*)d_in[15];
    float* out = (float*)d_out;

    const int N = in_sizes[0] / 3;
    const int E = in_sizes[1] / 2;
    const int Etot = E + N;
    const int nTiles = (N + 15) / 16;
    const int wmmaBlocks = (nTiles + (BLK / 32) - 1) / (BLK / 32);

    // workspace carve-up (floats)
    float* ws = (float*)d_ws;
    size_t off = 0;
    float* bufH   = ws + off; off += (size_t)N * 32;   // transformed features
    float* bufAgg = ws + off; off += (size_t)N * 32;   // aggregation accumulator
    float* bufX   = ws + off; off += (size_t)N * 32;   // layer output / next input
    float* alsA   = ws + off; off += (size_t)N * 2;
    float* aldA   = ws + off; off += (size_t)N * 2;
    float* mArr   = ws + off; off += (size_t)N * 2;
    float* sArr   = ws + off; off += (size_t)N * 2;
    float* elog   = ws + off; off += (size_t)Etot * 2;
    float* w3p    = ws + off; off += 512;
    (void)ws_size; (void)n_in; (void)out_size;

    // ---------------- Layer 1: H=2, C=16, K=3 (scalar transform, fused attn dots)
    fill_kernel<<<nblk((long long)N * 2),  BLK, 0, stream>>>(mArr, -1e30f, (long long)N * 2);
    fill_kernel<<<nblk((long long)N * 2),  BLK, 0, stream>>>(sArr, 0.0f,   (long long)N * 2);
    fill_kernel<<<nblk((long long)N * 32), BLK, 0, stream>>>(bufAgg, 0.0f, (long long)N * 32);
    linear1_kernel<<<nblk(N), BLK, 0, stream>>>(x, W1, as1, ad1, bufH, alsA, aldA, N);
    edge_logit_max_kernel<<<nblk(Etot), BLK, 0, stream>>>(ei, E, N, alsA, aldA, mArr, elog, 2);
    edge_exp_sum_kernel  <<<nblk(Etot), BLK, 0, stream>>>(ei, E, N, mArr, sArr, elog, 2);
    edge_scatter_kernel  <<<nblk((long long)Etot * 8), BLK, 0, stream>>>(ei, E, N, bufH, elog, sArr, bufAgg, 2, 16);
    finalize_kernel<<<nblk((long long)N * 32), BLK, 0, stream>>>(bufAgg, b1, bufX, N, 32);

    // ---------------- Layer 2: H=2, C=16, K=32 (WMMA transform)
    fill_kernel<<<nblk((long long)N * 2),  BLK, 0, stream>>>(mArr, -1e30f, (long long)N * 2);
    fill_kernel<<<nblk((long long)N * 2),  BLK, 0, stream>>>(sArr, 0.0f,   (long long)N * 2);
    fill_kernel<<<nblk((long long)N * 32), BLK, 0, stream>>>(bufAgg, 0.0f, (long long)N * 32);
    wmma_linear32_kernel<<<wmmaBlocks, BLK, 0, stream>>>(bufX, W2, bufH, nTiles, N);
    attn_kernel<<<nblk((long long)N * 2), BLK, 0, stream>>>(bufH, as2, ad2, alsA, aldA, N, 2, 16);
    edge_logit_max_kernel<<<nblk(Etot), BLK, 0, stream>>>(ei, E, N, alsA, aldA, mArr, elog, 2);
    edge_exp_sum_kernel  <<<nblk(Etot), BLK, 0, stream>>>(ei, E, N, mArr, sArr, elog, 2);
    edge_scatter_kernel  <<<nblk((long long)Etot * 8), BLK, 0, stream>>>(ei, E, N, bufH, elog, sArr, bufAgg, 2, 16);
    finalize_kernel<<<nblk((long long)N * 32), BLK, 0, stream>>>(bufAgg, b2, bufX, N, 32);

    // ---------------- Layer 3: H=1, C=8, K=32 (WMMA with zero-padded W3)
    fill_kernel<<<nblk((long long)N),     BLK, 0, stream>>>(mArr, -1e30f, (long long)N);
    fill_kernel<<<nblk((long long)N),     BLK, 0, stream>>>(sArr, 0.0f,   (long long)N);
    fill_kernel<<<nblk((long long)N * 8), BLK, 0, stream>>>(bufAgg, 0.0f, (long long)N * 8);
    padw3_kernel<<<2, BLK, 0, stream>>>(W3, w3p);
    wmma_linear8_kernel<<<wmmaBlocks, BLK, 0, stream>>>(bufX, w3p, bufH, nTiles, N);
    attn_kernel<<<nblk((long long)N), BLK, 0, stream>>>(bufH, as3, ad3, alsA, aldA, N, 1, 8);
    edge_logit_max_kernel<<<nblk(Etot), BLK, 0, stream>>>(ei, E, N, alsA, aldA, mArr, elog, 1);
    edge_exp_sum_kernel  <<<nblk(Etot), BLK, 0, stream>>>(ei, E, N, mArr, sArr, elog, 1);
    edge_scatter_kernel  <<<nblk((long long)Etot * 2), BLK, 0, stream>>>(ei, E, N, bufH, elog, sArr, bufAgg, 1, 8);
    finalize_kernel<<<nblk((long long)N * 8), BLK, 0, stream>>>(bufAgg, b3, bufX, N, 8);

    // ---------------- Output projection
    output_kernel<<<nblk(N), BLK, 0, stream>>>(bufX, Wo, bo, out, N);
}